// nodeselection_60163901883080
// MI455X (gfx1250) — compile-verified
//
#include <hip/hip_runtime.h>
#include <cstdint>
#include <cstddef>

// ---------------------------------------------------------------------------
// Problem constants (from the reference setup)
// ---------------------------------------------------------------------------
#define BDIM   16
#define TDIM   12
#define NNODE  4096
#define DDIM   64
#define MMEM   128
#define EDIM   128      // 2*D
#define TOPK   32
#define CHUNK  128      // N columns per LDS chunk
#define CH_STRIDE 132   // padded row stride (floats): 132 mod 64 banks = 4 ->
                        // full-wave conflict-free ds_load_b64, 16B row alignment
#define NCHUNK (NNODE / CHUNK)
#define THREADS 512
#define NWAVES (THREADS / 32)

#define LDS_CHUNK_FLOATS (CHUNK * CH_STRIDE)      // 16896 floats per buffer
#define LDS_LIST_ENTRIES (NWAVES * 16 * TOPK)     // 8192 entries
#define LDS_BYTES ((2 * LDS_CHUNK_FLOATS + 2 * LDS_LIST_ENTRIES) * 4)  // 200704 B

typedef float v2f __attribute__((ext_vector_type(2)));
typedef float v8f __attribute__((ext_vector_type(8)));

// Issue the async DMA of one 128-column chunk of nodevec3 (= concat(nv1,nv2))
// into an LDS buffer: 8 x GLOBAL_LOAD_ASYNC_TO_LDS_B128 per thread, ASYNCcnt-tracked.
__device__ __forceinline__ void issue_chunk_async(int c, float* buf,
                                                  const float* __restrict__ nv1,
                                                  const float* __restrict__ nv2,
                                                  size_t nvbase, int tid)
{
    #pragma unroll
    for (int it = 0; it < 8; ++it) {
        int slot = it * THREADS + tid;      // 0..4095 b128 slots (128 n x 32 slots)
        int nl = slot >> 5;
        int e  = (slot & 31) * 4;           // float index within the 128-wide row
        int n  = c * CHUNK + nl;
        const float* src = (e < DDIM)
            ? (nv1 + nvbase + (size_t)n * DDIM + e)
            : (nv2 + nvbase + (size_t)n * DDIM + (e - DDIM));
        // LDS aperture flat address truncates to allocation-relative byte offset
        unsigned loff = (unsigned)(size_t)(const void*)(buf + nl * CH_STRIDE + e);
        asm volatile("global_load_async_to_lds_b128 %0, %1, off"
                     :: "v"(loff), "v"(src) : "memory");
    }
}

// Comparator used throughout: (v1,i1) "greater" (v2,i2) iff v1>v2, tie -> lower index.
// This reproduces jax.lax.top_k's stable descending order exactly.

__global__ __launch_bounds__(THREADS)
void node_select_kernel(const float* __restrict__ nv1,
                        const float* __restrict__ nv2,
                        const float* __restrict__ emb,
                        float* __restrict__ out)
{
    extern __shared__ float smem[];
    float* ch0 = smem;                                // chunk buffer 0
    float* ch1 = smem + LDS_CHUNK_FLOATS;             // chunk buffer 1
    float* lv  = smem + 2 * LDS_CHUNK_FLOATS;         // per-wave top-k values
    int*   li  = (int*)(lv + LDS_LIST_ENTRIES);       // per-wave top-k indices

    const int tid  = threadIdx.x;
    const int w    = tid >> 5;
    const int lane = tid & 31;
    const int bt   = blockIdx.x >> 1;                 // b*T + t
    const int half = blockIdx.x & 1;                  // which 64-row m half
    const int b_   = bt / TDIM;
    const int t_   = bt - b_ * TDIM;

    const int mt  = w & 3;           // m-tile (16 rows) within the half
    const int ng  = w >> 2;          // n-tile group (2 n-tiles per chunk)
    const int m0g = half * 64 + mt * 16;

    // ---- init per-wave sorted lists: all (-inf, INT_MAX) ----
    #pragma unroll
    for (int r = 0; r < 16; ++r) {
        lv[(w * 16 + r) * TOPK + lane] = -INFINITY;
        li[(w * 16 + r) * TOPK + lane] = 0x7fffffff;
    }

    // ---- cache A (node_embeddings rows m0g..m0g+15, K=0..127) in WMMA layout ----
    // 16x4 f32 A layout: lane<16 -> (K=k0, k0+1); lane>=16 -> (K=k0+2, k0+3), M = lane&15
    const int am   = m0g + (lane & 15);
    const int koff = (lane >> 4) << 1;                // 0 or 2
    v2f aF[32];
    #pragma unroll
    for (int kk = 0; kk < 32; ++kk)
        aF[kk] = *(const v2f*)(emb + (size_t)am * EDIM + kk * 4 + koff);

    const size_t nvbase = (size_t)bt * NNODE * DDIM;

    // ---- software pipeline: async-DMA chunk c+1 while WMMA-ing chunk c ----
    issue_chunk_async(0, ch0, nv1, nv2, nvbase, tid);

    for (int c = 0; c < NCHUNK; ++c) {
        float* cur = (c & 1) ? ch1 : ch0;
        float* nxt = (c & 1) ? ch0 : ch1;
        if (c + 1 < NCHUNK) {
            issue_chunk_async(c + 1, nxt, nv1, nv2, nvbase, tid);
            // allow next chunk's 8 in flight; current chunk's 8 (in-order) are done
            asm volatile("s_wait_asynccnt 0x8" ::: "memory");
        } else {
            asm volatile("s_wait_asynccnt 0x0" ::: "memory");
        }
        __syncthreads();   // everyone's writes to `cur` are now visible

        #pragma unroll
        for (int tn = 0; tn < 2; ++tn) {
            const int nt = ng * 2 + tn;
            // ---- 16x16 tile GEMM, K=128, exact fp32 via v_wmma_f32_16x16x4_f32 ----
            // B 4x16 layout mirror of A: lane<16 -> rows K=k0,k0+1 at N=lane;
            // lane>=16 -> rows K=k0+2,k0+3 at N=lane-16. B[k][n] = nodevec3[n][k].
            v8f acc = {};
            const float* bbase = cur + (nt * 16 + (lane & 15)) * CH_STRIDE + koff;
            #pragma unroll
            for (int kk = 0; kk < 32; ++kk) {
                v2f bF = *(const v2f*)(bbase + kk * 4);
                acc = __builtin_amdgcn_wmma_f32_16x16x4_f32(
                    false, aF[kk], false, bF, (short)0, acc, false, false);
            }

            // ---- streaming top-k insert (per-wave lists, no races) ----
            const int nb = c * CHUNK + nt * 16 + (lane & 15);
            const bool upper = lane >= 16;
            #pragma unroll
            for (int r = 0; r < 16; ++r) {
                float v = acc[r & 7];     // D layout: row r<8 in lanes0-15, r>=8 in lanes16-31
                bool valid = (r < 8) ? (!upper) : upper;
                const int lb = (w * 16 + r) * TOPK;
                float th = lv[lb + TOPK - 1];
                unsigned long long mask = __ballot(valid && (v >= th));
                while (mask) {
                    int src = __builtin_ctzll(mask);  // increasing lane == increasing n
                    mask &= mask - 1;
                    float cv = __shfl(v, src);
                    int   cn = __shfl(nb, src);
                    float th2 = lv[lb + TOPK - 1];
                    int   ti2 = li[lb + TOPK - 1];
                    if (cv > th2 || (cv == th2 && cn < ti2)) {
                        // wave-parallel sorted insert with shift
                        float Lv = lv[lb + lane];
                        int   Li = li[lb + lane];
                        bool gtc = (Lv > cv) || (Lv == cv && Li < cn);
                        int pos = __builtin_popcountll(__ballot(gtc));
                        float sLv = __shfl_up(Lv, 1);
                        int   sLi = __shfl_up(Li, 1);
                        float nvv = (lane < pos) ? Lv : ((lane == pos) ? cv : sLv);
                        int   nii = (lane < pos) ? Li : ((lane == pos) ? cn : sLi);
                        lv[lb + lane] = nvv;
                        li[lb + lane] = nii;
                    }
                }
            }
        }
        __syncthreads();   // all reads of `cur` done before it is refilled next iter
    }

    // ---- merge the 4 per-wave lists per row, then write outputs + gather ----
    const size_t S1   = (size_t)BDIM * TDIM * MMEM * TOPK * DDIM;   // sel1 elements
    const size_t IDXN = (size_t)BDIM * TDIM * MMEM * TOPK;          // index-array elements

    for (int j = 0; j < 4; ++j) {
        const int r   = w * 4 + j;           // local row 0..63
        const int rmt = r >> 4, r16 = r & 15;
        int base0 = (rmt * 16 + r16) * TOPK;
        float Av = lv[base0 + lane];
        int   Ai = li[base0 + lane];
        #pragma unroll
        for (int s = 1; s < 4; ++s) {
            int bs = ((rmt + 4 * s) * 16 + r16) * TOPK;
            float Bv = lv[bs + (TOPK - 1 - lane)];
            int   Bi = li[bs + (TOPK - 1 - lane)];
            // desc ++ reversed-desc is bitonic; elementwise max keeps top-32 (bitonic)
            bool ag = (Av > Bv) || (Av == Bv && Ai < Bi);
            float Cv = ag ? Av : Bv;
            int   Ci = ag ? Ai : Bi;
            // 5-stage bitonic merge -> fully sorted descending
            #pragma unroll
            for (int d = 16; d >= 1; d >>= 1) {
                float Ov = __shfl_xor(Cv, d);
                int   Oi = __shfl_xor(Ci, d);
                bool gt = (Cv > Ov) || (Cv == Ov && Ci < Oi);
                bool keepmax = (lane & d) == 0;
                bool take = keepmax ? (!gt) : gt;
                Cv = take ? Ov : Cv;
                Ci = take ? Oi : Ci;
            }
            Av = Cv; Ai = Ci;
        }

        const int m = half * 64 + r;
        const size_t prow = ((size_t)bt * MMEM + m) * TOPK;
        const size_t p = prow + lane;        // k = lane
        out[2 * S1 + p]            = (float)b_;     // batch_indices
        out[2 * S1 + IDXN + p]     = (float)t_;     // time_indices
        out[2 * S1 + 2 * IDXN + p] = (float)Ai;     // indices

        // gather sel1/sel2: 64 floats per (m,k) from L2-hot nodevec rows
        #pragma unroll 1
        for (int k = 0; k < TOPK; ++k) {
            int gi = __shfl(Ai, k);
            const v2f* s1 = (const v2f*)(nv1 + nvbase + (size_t)gi * DDIM);
            const v2f* s2 = (const v2f*)(nv2 + nvbase + (size_t)gi * DDIM);
            v2f x1 = s1[lane];
            v2f x2 = s2[lane];
            *(v2f*)(out +       (prow + k) * DDIM + lane * 2) = x1;
            *(v2f*)(out + S1 +  (prow + k) * DDIM + lane * 2) = x2;
        }
    }
}

extern "C" void kernel_launch(void* const* d_in, const int* in_sizes, int n_in,
                              void* d_out, int out_size, void* d_ws, size_t ws_size,
                              hipStream_t stream) {
    (void)in_sizes; (void)n_in; (void)out_size; (void)d_ws; (void)ws_size;
    const float* nv1 = (const float*)d_in[0];
    const float* nv2 = (const float*)d_in[1];
    const float* emb = (const float*)d_in[2];
    // d_in[3] is topk == 32, compiled in.
    float* out = (float*)d_out;

    // allow >64KB dynamic LDS (CDNA5 WGP has 320KB); deterministic, capture-safe
    (void)hipFuncSetAttribute((const void*)node_select_kernel,
                              hipFuncAttributeMaxDynamicSharedMemorySize, LDS_BYTES);

    dim3 grid(BDIM * TDIM * 2);   // (b,t) x m-half = 384 workgroups
    dim3 block(THREADS);
    hipLaunchKernelGGL(node_select_kernel, grid, block, LDS_BYTES, stream,
                       nv1, nv2, emb, out);
}